// CPC_R_20323785245020
// MI455X (gfx1250) — compile-verified
//
#include <hip/hip_runtime.h>
#include <hip/hip_bf16.h>
#include <math.h>

// ---------------------------------------------------------------------------
// LM pose refinement on MI455X (gfx1250, wave32).
//
// Per point build augmented rows [Jx(6)|ex|0...], [Jy(6)|ey|0...].  Then
// Sum_r r^T r (7x7 block) = [JtJ, Jte; eJ, e^Te], computed as a SYRK with
// V_WMMA_F32_16X16X4_F32 (A 16x4 = chunk^T, B 4x16 = chunk, K=4 residual
// rows per instruction).  ISA layout (CDNA5 ISA, 32-bit A 16x4): lanes 0-15
// hold (M=lane,K=0/1) in VGPR0/1, lanes 16-31 hold K=2/3; B mirrors it
// (VGPR j, half h -> K=2h+j, N=lane&15), so for the symmetric product the
// per-lane A and B register contents are identical: one v2f feeds both.
// Two accumulators (even/odd chunks) break the D->C dependency chain.
// ---------------------------------------------------------------------------

typedef __attribute__((ext_vector_type(2))) float v2f;
typedef __attribute__((ext_vector_type(8))) float v8f;

// workspace layout (floats)
#define WS_PARAMS 0   // 6: r0 r1 r2 t0 t1 t2
#define WS_LAM    6
#define WS_MSE    7
#define WS_R      8   // 9: rotation, row major
#define WS_M      17  // 27: M0,M1,M2 (d cam/d r_i = M_i * (R p)), row major
#define WS_S      48  // 49: accumulated 7x7 augmented normal-equation block

__device__ __forceinline__ float fast_rcp(float x) {
  float r = __builtin_amdgcn_rcpf(x);       // v_rcp_f32 (~1 ulp)
  return r * (2.0f - x * r);                // one Newton step
}

// Rodrigues rotation + Gallego-Yezzi derivative matrices M_i.
__device__ inline void compute_RM(float r0, float r1, float r2, float* ws) {
  float th2 = r0 * r0 + r1 * r1 + r2 * r2;
  float th  = sqrtf(th2);
  float c = cosf(th), s = sinf(th);
  float it = 1.0f / th;
  float ux = r0 * it, uy = r1 * it, uz = r2 * it;
  float omc = 1.0f - c;
  float R[3][3];
  R[0][0] = c + omc * ux * ux;      R[0][1] = omc * ux * uy - s * uz; R[0][2] = omc * ux * uz + s * uy;
  R[1][0] = omc * uy * ux + s * uz; R[1][1] = c + omc * uy * uy;      R[1][2] = omc * uy * uz - s * ux;
  R[2][0] = omc * uz * ux - s * uy; R[2][1] = omc * uz * uy + s * ux; R[2][2] = c + omc * uz * uz;
  #pragma unroll
  for (int i = 0; i < 3; i++)
    #pragma unroll
    for (int j = 0; j < 3; j++) ws[WS_R + 3 * i + j] = R[i][j];

  float rv[3] = {r0, r1, r2};
  float it2 = 1.0f / th2;
  #pragma unroll
  for (int i = 0; i < 3; i++) {
    // c_i = e_i - R[:,i];  k = r x c_i;  M_i = (r_i*skew(r) + skew(k)) / th2
    float c0 = (i == 0 ? 1.0f : 0.0f) - R[0][i];
    float c1 = (i == 1 ? 1.0f : 0.0f) - R[1][i];
    float c2 = (i == 2 ? 1.0f : 0.0f) - R[2][i];
    float k0 = r1 * c2 - r2 * c1;
    float k1 = r2 * c0 - r0 * c2;
    float k2 = r0 * c1 - r1 * c0;
    float ri = rv[i];
    float* M = ws + WS_M + 9 * i;
    M[0] = 0.0f;                  M[1] = (-ri * r2 - k2) * it2; M[2] = ( ri * r1 + k1) * it2;
    M[3] = ( ri * r2 + k2) * it2; M[4] = 0.0f;                  M[5] = (-ri * r0 - k0) * it2;
    M[6] = (-ri * r1 - k1) * it2; M[7] = ( ri * r0 + k0) * it2; M[8] = 0.0f;
  }
}

__global__ void lm_init_kernel(const float* __restrict__ init_r,
                               const float* __restrict__ init_t,
                               float* __restrict__ ws) {
  if (threadIdx.x == 0 && blockIdx.x == 0) {
    float p[6] = {init_r[0], init_r[1], init_r[2], init_t[0], init_t[1], init_t[2]};
    #pragma unroll
    for (int i = 0; i < 6; i++) ws[WS_PARAMS + i] = p[i];
    ws[WS_LAM] = -1.0f;  // sentinel: lam initialized on first solve
    ws[WS_MSE] = 0.0f;
    compute_RM(p[0], p[1], p[2], ws);
    for (int i = 0; i < 49; i++) ws[WS_S + i] = 0.0f;
  }
}

// Big kernel: per-point Jacobians -> WMMA SYRK accumulation -> 49 atomics/block.
__global__ void __launch_bounds__(256)
lm_accum_kernel(const float* __restrict__ p3, const float* __restrict__ p2,
                const float* __restrict__ fo, const float* __restrict__ ce,
                float* __restrict__ ws, int N) {
  // 16-wide rows: cols 8..15 are zero-filled once, so WMMA operand loads for
  // param index >= 8 need no lane predication.  32 KB of the 320 KB/WGP LDS.
  __shared__ float stage[8][64][16];
  __shared__ float Ssh[49];

  const int tid  = threadIdx.x;
  const int wave = tid >> 5;
  const int lane = tid & 31;
  const int half = lane >> 4;   // lane-half selects K pair
  const int pr   = lane & 15;   // param (M/N) index

  // uniform per-iteration constants
  const float fx = fo[0], fy = fo[1], cx = ce[0], cy = ce[1];
  const float tx = ws[WS_PARAMS + 3], ty = ws[WS_PARAMS + 4], tz = ws[WS_PARAMS + 5];
  float Rm[9], Mm[27];
  #pragma unroll
  for (int i = 0; i < 9; i++) Rm[i] = ws[WS_R + i];
  #pragma unroll
  for (int i = 0; i < 27; i++) Mm[i] = ws[WS_M + i];

  if (tid < 49) Ssh[tid] = 0.0f;
  // zero the invariant pad columns of this lane's two rows (wave-private slab)
  #pragma unroll
  for (int e = 8; e < 16; e++) {
    stage[wave][2 * lane + 0][e] = 0.0f;
    stage[wave][2 * lane + 1][e] = 0.0f;
  }
  __syncthreads();

  v8f acc0 = {0.0f, 0.0f, 0.0f, 0.0f, 0.0f, 0.0f, 0.0f, 0.0f};
  v8f acc1 = {0.0f, 0.0f, 0.0f, 0.0f, 0.0f, 0.0f, 0.0f, 0.0f};

  const int stride = gridDim.x * blockDim.x;
  for (int base = blockIdx.x * blockDim.x; base < N; base += stride) {
    const int i = base + tid;
    float jx[8], jy[8];
    if (i < N) {
      const float px = p3[3 * i], py = p3[3 * i + 1], pz = p3[3 * i + 2];
      const float w0 = Rm[0] * px + Rm[1] * py + Rm[2] * pz;
      const float w1 = Rm[3] * px + Rm[4] * py + Rm[5] * pz;
      const float w2 = Rm[6] * px + Rm[7] * py + Rm[8] * pz;
      const float camx = w0 + tx, camy = w1 + ty, camz = w2 + tz;
      const float iz  = fast_rcp(camz);
      const float ax  = fx * iz, by = fy * iz;
      const float gxz = -ax * camx * iz;
      const float gyz = -by * camy * iz;
      const float ex  = fx * camx * iz + cx - p2[2 * i];
      const float ey  = fy * camy * iz + cy - p2[2 * i + 1];
      #pragma unroll
      for (int k = 0; k < 3; k++) {
        const float* M = Mm + 9 * k;
        const float d0 = M[0] * w0 + M[1] * w1 + M[2] * w2;
        const float d1 = M[3] * w0 + M[4] * w1 + M[5] * w2;
        const float d2 = M[6] * w0 + M[7] * w1 + M[8] * w2;
        jx[k] = ax * d0 + gxz * d2;
        jy[k] = by * d1 + gyz * d2;
      }
      jx[3] = ax;   jx[4] = 0.0f; jx[5] = gxz; jx[6] = ex; jx[7] = 0.0f;
      jy[3] = 0.0f; jy[4] = by;   jy[5] = gyz; jy[6] = ey; jy[7] = 0.0f;
    } else {
      #pragma unroll
      for (int e = 0; e < 8; e++) { jx[e] = 0.0f; jy[e] = 0.0f; }
    }

    // stage this wave's 64 augmented rows (LDS ops are in-order per wave;
    // slab is wave-private, so no block barrier needed)
    #pragma unroll
    for (int e = 0; e < 8; e++) {
      stage[wave][2 * lane + 0][e] = jx[e];
      stage[wave][2 * lane + 1][e] = jy[e];
    }

    // 16 WMMAs consume 4 residual rows each: acc += chunk^T * chunk.
    // Even/odd chunks go to separate accumulators for matrix-pipe ILP.
    #pragma unroll
    for (int ck = 0; ck < 16; ck += 2) {
      const int r0 = 4 * ck + 2 * half;
      v2f ab0, ab1;
      ab0.x = stage[wave][r0 + 0][pr];
      ab0.y = stage[wave][r0 + 1][pr];
      ab1.x = stage[wave][r0 + 4][pr];
      ab1.y = stage[wave][r0 + 5][pr];
      acc0 = __builtin_amdgcn_wmma_f32_16x16x4_f32(
          false, ab0, false, ab0, (short)0, acc0, false, false);
      acc1 = __builtin_amdgcn_wmma_f32_16x16x4_f32(
          false, ab1, false, ab1, (short)0, acc1, false, false);
    }
  }

  v8f acc = acc0 + acc1;

  // D layout: acc[v] = D[m][n], n = lane&15, m = v + 8*(lane>=16).
  // Live entries: m,n in [0,6] -> lanes 0..6, VGPRs 0..6.
  if (lane < 7) {
    #pragma unroll
    for (int v = 0; v < 7; v++) atomicAdd(&Ssh[v * 7 + lane], acc[v]);
  }
  __syncthreads();
  if (tid < 49) atomicAdd(&ws[WS_S + tid], Ssh[tid]);
}

// Tiny kernel: 6x6 LM solve, param update, constants for next iteration.
__global__ void lm_solve_kernel(float* __restrict__ ws, float* __restrict__ out,
                                float inv2N) {
  if (threadIdx.x != 0 || blockIdx.x != 0) return;
  float S[49];
  for (int i = 0; i < 49; i++) S[i] = ws[WS_S + i];

  float lam = ws[WS_LAM];
  if (lam < 0.0f) {
    float md = S[0];
    for (int d = 1; d < 6; d++) md = fmaxf(md, S[d * 7 + d]);
    lam = 1e-8f * md;
    ws[WS_LAM] = lam;
  }

  // augmented system  (JtJ + lam I | Jte)
  float A[6][7];
  for (int i = 0; i < 6; i++) {
    for (int j = 0; j < 6; j++) A[i][j] = S[i * 7 + j];
    A[i][i] += lam;
    A[i][6] = S[i * 7 + 6];
  }
  // Gauss-Jordan with partial pivoting
  for (int col = 0; col < 6; col++) {
    int piv = col;
    float best = fabsf(A[col][col]);
    for (int r = col + 1; r < 6; r++) {
      float v = fabsf(A[r][col]);
      if (v > best) { best = v; piv = r; }
    }
    if (piv != col)
      for (int j = 0; j < 7; j++) { float t = A[col][j]; A[col][j] = A[piv][j]; A[piv][j] = t; }
    float inv = 1.0f / A[col][col];
    for (int j = 0; j < 7; j++) A[col][j] *= inv;
    for (int r = 0; r < 6; r++) {
      if (r == col) continue;
      float f = A[r][col];
      for (int j = 0; j < 7; j++) A[r][j] -= f * A[col][j];
    }
  }

  float p[6];
  for (int i = 0; i < 6; i++) {
    p[i] = ws[WS_PARAMS + i] - A[i][6];  // params += (-solve)
    ws[WS_PARAMS + i] = p[i];
    out[i] = p[i];
  }
  const float mse = S[48] * inv2N;  // S[6][6] = sum err^2 over 2N residuals
  ws[WS_MSE] = mse;
  out[6] = mse;

  compute_RM(p[0], p[1], p[2], ws);        // constants for next accum
  for (int i = 0; i < 49; i++) ws[WS_S + i] = 0.0f;
}

extern "C" void kernel_launch(void* const* d_in, const int* in_sizes, int n_in,
                              void* d_out, int out_size, void* d_ws, size_t ws_size,
                              hipStream_t stream) {
  const float* p3 = (const float*)d_in[0];  // points3d [N,3]
  const float* p2 = (const float*)d_in[1];  // points2d [N,2]
  const float* ir = (const float*)d_in[2];  // initial rodrigues [3]
  const float* it = (const float*)d_in[3];  // initial tr [3]
  const float* fo = (const float*)d_in[4];  // focals [2]
  const float* ce = (const float*)d_in[5];  // centers [2]
  const int N = in_sizes[0] / 3;
  float* ws  = (float*)d_ws;
  float* out = (float*)d_out;

  lm_init_kernel<<<1, 32, 0, stream>>>(ir, it, ws);

  const int grid = 1024;               // 256K threads, 8 grid-stride passes
  const float inv2N = 1.0f / (2.0f * (float)N);
  const int n_iters = 8;               // reference constant
  for (int iter = 0; iter < n_iters; iter++) {
    lm_accum_kernel<<<grid, 256, 0, stream>>>(p3, p2, fo, ce, ws, N);
    lm_solve_kernel<<<1, 32, 0, stream>>>(ws, out, inv2N);
  }
}